// TopnMSELoss_44787918962929
// MI455X (gfx1250) — compile-verified
//
#include <hip/hip_runtime.h>
#include <hip/hip_bf16.h>

typedef __attribute__((ext_vector_type(2))) float v2f;
typedef __attribute__((ext_vector_type(8))) float v8f;

#define BROWS 512
#define TOPK 128u
#define SEL_THREADS 512
#define NWAVES (SEL_THREADS / 32)
#define TIE_CAP 256

// Monotonic unsigned key: key(a) < key(b)  <=>  a < b  (for non-NaN floats)
__device__ __forceinline__ unsigned key_of(float f) {
    unsigned u = __float_as_uint(f);
    return (u & 0x80000000u) ? ~u : (u | 0x80000000u);
}

// Wave32 sum of `part` across all 32 lanes, routed through V_WMMA_F32_16X16X4_F32.
// A = all ones (16x4) => C[m,n] = sum_k B[k,n]; every C row is identical, and
// sum over all n of one row equals the sum of ALL 64 B slots, independent of the
// hardware (k,n)<->(lane,vgpr) mapping. VGPR0 of C holds rows M=0 (lanes 0-15)
// and M=8 (lanes 16-31), so the full cross-lane sum of c[0] is 2x the total.
// Requires EXEC all-1s: call only in uniform control flow with all lanes live.
__device__ __forceinline__ float wave_sum_wmma(float part) {
    v2f a; a.x = 1.0f; a.y = 1.0f;      // 64 A slots, all 1.0
    v2f b; b.x = part; b.y = 0.0f;      // 64 B slots: 32 partials + 32 zeros
    v8f c = {};
    c = __builtin_amdgcn_wmma_f32_16x16x4_f32(
            /*neg_a=*/false, a, /*neg_b=*/false, b,
            /*c_mod=*/(short)0, c, /*reuse_a=*/false, /*reuse_b=*/false);
    float v = c[0];
#pragma unroll
    for (int off = 16; off >= 1; off >>= 1)
        v += __shfl_xor(v, off, 32);
    return v * 0.5f;                    // every lane holds the wave sum
}

// Kernel 1: colsum[j] = sum_i (s[i,j]-t[i,j])^2.  Coalesced: lanes cover
// adjacent columns, loop walks rows.
__global__ void __launch_bounds__(256) colsum_kernel(
        const float* __restrict__ s, const float* __restrict__ t,
        float* __restrict__ colsum, int n) {
    int j = blockIdx.x * blockDim.x + threadIdx.x;
    if (j >= n) return;
    const float* sp = s + j;
    const float* tp = t + j;
    float acc = 0.0f;
    for (int b = 0; b < BROWS; ++b) {
        float d = sp[(size_t)b * (size_t)n] - tp[(size_t)b * (size_t)n];
        acc = fmaf(d, d, acc);
    }
    colsum[j] = acc;
}

// Kernel 2: one workgroup per row. 4-pass 8-bit LDS radix select of the exact
// 32-bit key T of the 128th-smallest element, then one emit pass summing
// colsum[j] over selected indices (key<T plus lowest-index ties ==T).
__global__ void __launch_bounds__(SEL_THREADS) select_kernel(
        const float* __restrict__ s, const float* __restrict__ colsum,
        float* __restrict__ rowsum, int n) {
    __shared__ unsigned hist[256];
    __shared__ unsigned tieIdx[TIE_CAP];
    __shared__ unsigned tieCnt;
    __shared__ unsigned sPrefix, sKth;
    __shared__ float    tieSum;
    __shared__ float    warr[NWAVES];

    const int tid = threadIdx.x;
    const int row = blockIdx.x;
    const float* __restrict__ rowp = s + (size_t)row * (size_t)n;

    if (tid == 0) { sKth = TOPK; sPrefix = 0u; tieCnt = 0u; tieSum = 0.0f; }

    unsigned prefix = 0u;   // right-aligned bits already resolved
#pragma unroll
    for (int pass = 0; pass < 4; ++pass) {
        const int sh = 24 - 8 * pass;
        __syncthreads();                 // also covers the tid==0 init above
        if (tid < 256) hist[tid] = 0u;
        __syncthreads();
        for (int j = tid; j < n; j += SEL_THREADS) {
            unsigned key = key_of(rowp[j]);
            // 64-bit shift keeps pass 0 (shift by 32) well-defined: 0 == prefix(0)
            if (((unsigned long long)key >> (sh + 8)) == (unsigned long long)prefix)
                atomicAdd(&hist[(key >> sh) & 0xFFu], 1u);
        }
        __syncthreads();
        if (tid == 0) {
            unsigned kth = sKth, cum = 0u, bin = 0u;
            for (bin = 0u; bin < 256u; ++bin) {
                unsigned cnew = cum + hist[bin];
                if (cnew >= kth) break;
                cum = cnew;
            }
            sKth    = kth - cum;                 // rank within chosen bin
            sPrefix = (sPrefix << 8) | bin;
        }
        __syncthreads();
        prefix = sPrefix;
    }

    const unsigned T = sPrefix;   // exact key of the 128th smallest
    const unsigned m = sKth;      // how many ==T elements to include (>=1)

    // Emit pass: accumulate colsum for strict members, collect tie indices.
    float part = 0.0f;
    for (int j = tid; j < n; j += SEL_THREADS) {
        unsigned key = key_of(rowp[j]);
        if (key < T) {
            part += colsum[j];
        } else if (key == T) {
            unsigned p = atomicAdd(&tieCnt, 1u);
            if (p < TIE_CAP) tieIdx[p] = (unsigned)j;
        }
    }
    __syncthreads();
    if (tid == 0) {
        unsigned cnt = tieCnt; if (cnt > TIE_CAP) cnt = TIE_CAP;
        // insertion sort ascending (cnt is tiny; indices are unique) -> deterministic
        for (unsigned i = 1u; i < cnt; ++i) {
            unsigned v = tieIdx[i]; int k = (int)i - 1;
            while (k >= 0 && tieIdx[k] > v) { tieIdx[k + 1] = tieIdx[k]; --k; }
            tieIdx[k + 1] = v;
        }
        unsigned take = (m < cnt) ? m : cnt;
        float ts = 0.0f;
        for (unsigned i = 0u; i < take; ++i) ts += colsum[tieIdx[i]];
        tieSum = ts;
    }
    __syncthreads();
    if (tid == 0) part += tieSum;

    // Deterministic block reduction: WMMA wave sum, then fixed-order combine.
    float wsum = wave_sum_wmma(part);
    int lane = tid & 31, wid = tid >> 5;
    if (lane == 0) warr[wid] = wsum;
    __syncthreads();
    if (tid == 0) {
        float tot = 0.0f;
        for (int w = 0; w < NWAVES; ++w) tot += warr[w];
        rowsum[row] = tot;
    }
}

// Kernel 3: reduce 512 row sums to the scalar output (single block, fixed order).
__global__ void __launch_bounds__(SEL_THREADS) final_kernel(
        const float* __restrict__ rowsum, float* __restrict__ out) {
    __shared__ float warr[NWAVES];
    int tid = threadIdx.x;
    float part = rowsum[tid];           // 512 threads, 512 rows
    float wsum = wave_sum_wmma(part);
    int lane = tid & 31, wid = tid >> 5;
    if (lane == 0) warr[wid] = wsum;
    __syncthreads();
    if (tid == 0) {
        float tot = 0.0f;
        for (int w = 0; w < NWAVES; ++w) tot += warr[w];
        out[0] = tot;
    }
}

extern "C" void kernel_launch(void* const* d_in, const int* in_sizes, int n_in,
                              void* d_out, int out_size, void* d_ws, size_t ws_size,
                              hipStream_t stream) {
    const float* student = (const float*)d_in[0];
    const float* teacher = (const float*)d_in[1];
    const int n = in_sizes[0] / BROWS;          // 50257 columns

    float* colsum = (float*)d_ws;               // n floats
    float* rowsum = colsum + n;                 // BROWS floats
    float* out    = (float*)d_out;

    const int cthreads = 256;
    const int cblocks  = (n + cthreads - 1) / cthreads;
    colsum_kernel<<<cblocks, cthreads, 0, stream>>>(student, teacher, colsum, n);
    select_kernel<<<BROWS, SEL_THREADS, 0, stream>>>(student, colsum, rowsum, n);
    final_kernel<<<1, SEL_THREADS, 0, stream>>>(rowsum, out);
}